// FSI_Block_35785667510478
// MI455X (gfx1250) — compile-verified
//
#include <hip/hip_runtime.h>
#include <math.h>

#define Bn 8
#define Hn 768
#define Wn 768
#define HWn (Hn * Wn)
#define STEPS 6

typedef __attribute__((ext_vector_type(2))) float v2f;
typedef __attribute__((ext_vector_type(8))) float v8f;

// flat LDS offset of tap t inside the 9x130 patch: row = ic*3+dy, col shift dx
#define TOFF(t) (((((t) / 9) * 3 + (((t) % 9) / 3)) * 130) + ((t) % 3))

// ---------------------------------------------------------------------------
// Kernel 1: 3x3 conv (3 in ch -> 5 out ch, zero 'SAME' pad) fused with the
// diffusion-tensor derivation. Implicit GEMM on V_WMMA_F32_16X16X4_F32:
//   D[16 pix x 16 oc] = sum over 7 K-chunks of A[16x4 taps] * B[4 taps x 16 oc]
// Weights staged zero-padded (28 taps x 16 ocs) in LDS, so tap 27 / oc>=5
// contribute exact zeros and no masking is needed on the A side.
// Output: packed float4(a, b, c, alpha) per (batch, y, x).
// ---------------------------------------------------------------------------
__global__ __launch_bounds__(256) void conv_coeff_kernel(
    const float* __restrict__ u, const float* __restrict__ f,
    const float* __restrict__ w, const float* __restrict__ bias,
    const float* __restrict__ contrast_p, float4* __restrict__ coeff)
{
  __shared__ float patch[9 * 130];   // [ic*3+ry][cx], rows y-1..y+1, cols x0-1..x0+128
  __shared__ float wlds[28 * 16];    // [tap][oc], zero padded
  __shared__ float xls[8][5][16];    // per-wave conv outputs (5 used ocs)

  const int x0  = blockIdx.x * 128;
  const int y   = blockIdx.y;
  const int b   = blockIdx.z;
  const int tid = threadIdx.x;

  // ---- stage input patch (zero padded, SAME conv) ----
  for (int li = tid; li < 9 * 130; li += 256) {
    int ic  = li / 390;
    int rem = li - ic * 390;
    int ry  = rem / 130;
    int cx  = rem - ry * 130;
    int gy  = y - 1 + ry;
    int gx  = x0 - 1 + cx;
    float v = 0.0f;
    if (gy >= 0 && gy < Hn && gx >= 0 && gx < Wn) {
      const float* plane = (ic < 2) ? (u + (size_t)(b * 2 + ic) * HWn)
                                    : (f + (size_t)b * HWn);
      v = plane[gy * Wn + gx];
    }
    patch[li] = v;
  }
  // ---- stage zero-padded weight matrix: wlds[t][n] = w[n*27+t] ----
  for (int li = tid; li < 28 * 16; li += 256) {
    const int t = li >> 4;
    const int n = li & 15;
    wlds[li] = (t < 27 && n < 5) ? w[n * 27 + t] : 0.0f;
  }
  __syncthreads();

  const int wv    = tid >> 5;        // wave id 0..7 (wave32)
  const int lane  = tid & 31;
  const int mOrN  = lane & 15;       // M (pixel) for A, N (out ch) for B
  const int hi    = lane >> 4;       // lane half selects K pair
  const int px    = wv * 16 + mOrN;  // pixel within 128-wide segment
  const int bbase = hi * 32 + mOrN;  // B-fragment per-lane base (float index)

  v8f acc = {};
  #pragma unroll
  for (int kc = 0; kc < 7; ++kc) {
    v2f afrag, bfrag;
    #pragma unroll
    for (int r = 0; r < 2; ++r) {
      const int tl = 4 * kc + r;                                  // lanes 0-15 tap
      const int th = (4 * kc + 2 + r > 26) ? 26 : (4 * kc + 2 + r); // lanes 16-31 tap (27->clamped, B row is 0)
      const int ol = TOFF(tl);
      const int oh = TOFF(th);
      afrag[r] = patch[px + (hi ? oh : ol)];    // cndmask of 2 immediates + ds_load
      bfrag[r] = wlds[bbase + tl * 16];         // ds_load with immediate offset
    }
    // 8 args: (neg_a, A, neg_b, B, c_mod, C, reuse_a, reuse_b)
    acc = __builtin_amdgcn_wmma_f32_16x16x4_f32(false, afrag, false, bfrag,
                                                (short)0, acc, false, false);
  }

  // D layout: lane holds column N=mOrN, rows M = r + 8*hi
  if (mOrN < 5) {
    #pragma unroll
    for (int r = 0; r < 8; ++r)
      xls[wv][mOrN][r + 8 * hi] = acc[r];
  }
  __syncthreads();

  // ---- per-pixel coefficient derivation ----
  if (tid < 128) {
    float xv[5];
    #pragma unroll
    for (int ch = 0; ch < 5; ++ch)
      xv[ch] = xls[tid >> 4][ch][tid & 15] + bias[ch];

    float v1 = xv[0], v2 = xv[1];
    float nrm = fmaxf(sqrtf(v1 * v1 + v2 * v2), 1e-12f);
    v1 /= nrm; v2 /= nrm;
    const float ct   = *contrast_p;
    const float den  = ct * ct + 1e-8f;
    const float lam1 = 1.0f / (1.0f + xv[2] * xv[2] / den);
    const float lam2 = 1.0f / (1.0f + xv[3] * xv[3] / den);
    const float a    = v1 * v1 * lam1 + v2 * v2 * lam2;
    const float c    = lam1 + lam2 - a;
    const float bb   = v1 * v2 * (lam1 - lam2);
    const float alph = 0.5f / (1.0f + expf(-xv[4]));
    coeff[((size_t)b * Hn + y) * Wn + x0 + tid] = make_float4(a, bb, c, alph);
  }
}

// ---------------------------------------------------------------------------
// Diffusion step, both u-channels fused per thread so the (channel-less)
// coefficient field is loaded once and applied twice.
//   ux,uy : central differences with edge clamp on u
//   j1=a*ux+b*uy, j2=b*ux+c*uy ; div = central diff of j1 (x) + j2 (y) with
//   zero boundary ; out = ai*(u + alpha*div) + (1-ai)*u_prev
// ---------------------------------------------------------------------------
__device__ __forceinline__ float d_x(const float* p, int y, int x) {
  const int xm = x > 0 ? x - 1 : 0;
  const int xp = x < Wn - 1 ? x + 1 : Wn - 1;
  return 0.5f * (p[y * Wn + xp] - p[y * Wn + xm]);
}
__device__ __forceinline__ float d_y(const float* p, int y, int x) {
  const int ym = y > 0 ? y - 1 : 0;
  const int yp = y < Hn - 1 ? y + 1 : Hn - 1;
  return 0.5f * (p[yp * Wn + x] - p[ym * Wn + x]);
}

__global__ __launch_bounds__(256) void diff_step_kernel(
    const float* __restrict__ cur, const float* __restrict__ prev,
    const float4* __restrict__ coeff, float* __restrict__ out,
    const float* __restrict__ alphas, int step)
{
  const int idx = blockIdx.x * 256 + threadIdx.x;  // over B*H*W pixels
  if (idx >= Bn * HWn) return;

  const int x = idx % Wn;
  const int y = (idx / Wn) % Hn;
  const int b = idx / HWn;

  const float4* cf = coeff + (size_t)b * HWn;
  const float4 kc  = cf[y * Wn + x];
  const float  ai  = alphas[step];

  #pragma unroll
  for (int ch = 0; ch < 2; ++ch) {
    const size_t plane = (size_t)(b * 2 + ch) * HWn;
    const float* p = cur + plane;

    float j1m = 0.f, j1p = 0.f, j2m = 0.f, j2p = 0.f;
    if (x > 0) {
      const float4 k = cf[y * Wn + (x - 1)];
      j1m = k.x * d_x(p, y, x - 1) + k.y * d_y(p, y, x - 1);
    }
    if (x < Wn - 1) {
      const float4 k = cf[y * Wn + (x + 1)];
      j1p = k.x * d_x(p, y, x + 1) + k.y * d_y(p, y, x + 1);
    }
    if (y > 0) {
      const float4 k = cf[(y - 1) * Wn + x];
      j2m = k.y * d_x(p, y - 1, x) + k.z * d_y(p, y - 1, x);
    }
    if (y < Hn - 1) {
      const float4 k = cf[(y + 1) * Wn + x];
      j2p = k.y * d_x(p, y + 1, x) + k.z * d_y(p, y + 1, x);
    }
    const float div   = 0.5f * (j1p - j1m) + 0.5f * (j2p - j2m);
    const float ustep = p[y * Wn + x] + kc.w * div;
    out[plane + y * Wn + x] = ai * ustep + (1.0f - ai) * prev[plane + y * Wn + x];
  }
}

// ---------------------------------------------------------------------------
extern "C" void kernel_launch(void* const* d_in, const int* in_sizes, int n_in,
                              void* d_out, int out_size, void* d_ws, size_t ws_size,
                              hipStream_t stream)
{
  (void)in_sizes; (void)n_in; (void)out_size; (void)ws_size;

  const float* u        = (const float*)d_in[0];
  const float* f        = (const float*)d_in[1];
  const float* w        = (const float*)d_in[2];
  const float* bias     = (const float*)d_in[3];
  const float* contrast = (const float*)d_in[4];
  const float* alphas   = (const float*)d_in[5];

  float4* coeff = (float4*)d_ws;
  float*  ws1   = (float*)((char*)d_ws + (size_t)Bn * HWn * sizeof(float4));
  float*  ws2   = ws1 + (size_t)Bn * 2 * HWn;
  float*  outp  = (float*)d_out;

  dim3 cgrid(Wn / 128, Hn, Bn);
  conv_coeff_kernel<<<cgrid, 256, 0, stream>>>(u, f, w, bias, contrast, coeff);

  const int N = Bn * HWn;               // one thread per pixel, channels fused
  const int blocks = (N + 255) / 256;

  // loop state rotation: u_prev/u start as the input; in-place overwrite of
  // `prev` is safe (prev is only read at the owning pixel before the write).
  const float* curp[STEPS] = { u,   ws1, ws2, ws1, ws2, ws1 };
  const float* prvp[STEPS] = { u,   u,   ws1, ws2, ws1, ws2 };
  float*       nxtp[STEPS] = { ws1, ws2, ws1, ws2, ws1, outp };

  for (int s = 0; s < STEPS; ++s)
    diff_step_kernel<<<blocks, 256, 0, stream>>>(curp[s], prvp[s], coeff,
                                                 nxtp[s], alphas, s);
}